// VMDecomposition_8735963480351
// MI455X (gfx1250) — compile-verified
//
#include <hip/hip_runtime.h>
#include <stdint.h>

// ---------------------------------------------------------------------------
// VM decomposition (TensoRF): out[p,c] = sum_i bilinear(plane_i)[p,c] * lc_i[c]
// where lc_i[c] = 0.5*(line_i[c,255] + line_i[c,256])  (line sample collapses
// to a per-channel constant because gy==0 -> y=255.5 and W=2 columns are dup).
//
// Strategy (MI455X):
//  * 96 MB of planes fit in 192 MB L2 -> gather-line-traffic bound, not HBM.
//  * One-time transpose to channel-fastest layout [512][512][32] in d_ws so
//    each corner gather = exactly one 128B line per wave (lane == channel,
//    wave32 matches R=32 exactly).
//  * Transpose uses the Tensor Data Mover: one tensor_load_to_lds per
//    (y, 128-x) tile, 2D descriptor with stride 512*512 elems between channel
//    lines, pad_enable => LDS row stride 129 dwords (bank-conflict-free
//    transposed reads). Completion via s_wait_tensorcnt.
// ---------------------------------------------------------------------------

#define R_CH 32
#define RES 512
#define NPTS (2048 * 256)                         // 524288 points
#define CH_STRIDE (RES * RES)                     // 262144 elems between channels
#define PLANE_ELEMS (R_CH * RES * RES)            // 8388608 floats / plane
#define WS_NEEDED (3ull * PLANE_ELEMS * 4ull)     // 96 MB transposed planes

#define XB 128          // x elements per transpose tile
#define SPAD 129        // padded LDS row stride (dwords)
#define PPW 8           // points per wave in the gather kernel

typedef unsigned int v4u __attribute__((ext_vector_type(4)));
typedef int          v8i __attribute__((ext_vector_type(8)));
typedef int          v4i __attribute__((ext_vector_type(4)));

// ---------------------------------------------------------------------------
// Transpose kernel: [32][512][512] -> [512][512][32], tiled (y, 128 x) per WG.
// ---------------------------------------------------------------------------
__global__ __launch_bounds__(256) void vm_transpose_planes(
    const float* __restrict__ p0, const float* __restrict__ p1,
    const float* __restrict__ p2, float* __restrict__ dst)
{
  __shared__ float stage[R_CH * SPAD];   // 4128 dwords = 16512 B (incl. pads)
  const int tid = threadIdx.x;
  const int xb  = blockIdx.x * XB;
  const int y   = blockIdx.y;
  const int pl  = blockIdx.z;
  const float* src = (pl == 0) ? p0 : (pl == 1) ? p1 : p2;
  float* out = dst + (size_t)pl * PLANE_ELEMS + (size_t)(y * RES + xb) * R_CH;

#if __has_builtin(__builtin_amdgcn_tensor_load_to_lds) && \
    __has_builtin(__builtin_amdgcn_s_wait_tensorcnt)
  if (tid < 32) {  // wave 0 issues one TDM tile load (EXEC ignored by TDM)
    unsigned long long ga =
        (unsigned long long)(uintptr_t)(src + (size_t)y * RES + xb);
    unsigned ldsOff = (unsigned)(uintptr_t)&stage[0];
    // D# group 0: count=1 (load), lds_addr, global_addr[56:0], type=2
    v4u g0 = { 1u,
               ldsOff,
               (unsigned)(ga & 0xffffffffu),
               (unsigned)((ga >> 32) & 0x01ffffffu) | (2u << 30) };
    // D# group 1: data_size=4B(code2), pad_enable, pad_interval=128dw(code6),
    // pad_amount=1dw(code0); tensor viewed as 2D [32 ch x 262144],
    // tile = [tile_dim0=128 x, tile_dim1=32 ch], line stride = 262144 elems.
    v8i g1 = { (int)((2u << 16) | (1u << 20) | (6u << 22)), // flags
               (int)((CH_STRIDE & 0xffffu) << 16),          // tensor_dim0 lo16 (=0)
               (int)((CH_STRIDE >> 16) | (32u << 16)),      // dim0 hi16 | tensor_dim1 lo16
               (int)(128u << 16),                           // tensor_dim1 hi16 | tile_dim0
               (int)(32u),                                  // tile_dim1 | tile_dim2=0
               (int)(CH_STRIDE),                            // tensor_dim0_stride lo32
               0,                                           // stride0 hi16 | stride1 lo16
               0 };                                         // stride1 hi32
    v4i g2 = { 0, 0, 0, 0 };   // dims 2/3 unused (2D tile)
    v4i g3 = { 0, 0, 0, 0 };
    v8i gx = { 0, 0, 0, 0, 0, 0, 0, 0 };  // extra group (6-arg toolchain), zero
    __builtin_amdgcn_tensor_load_to_lds(g0, g1, g2, g3, gx, 0);
    __builtin_amdgcn_s_wait_tensorcnt(0);
  }
  __syncthreads();
#else
  // Manual staging fallback: coalesced b128 reads along x, padded LDS writes.
  {
    const int c  = tid >> 3;          // 0..31 channel
    const int xg = (tid & 7) * 16;    // 0..112
    const float* rowp = src + (size_t)c * CH_STRIDE + (size_t)y * RES + xb + xg;
#pragma unroll
    for (int k = 0; k < 4; ++k) {
      float4 v = *(const float4*)(rowp + 4 * k);
      float* s = &stage[c * SPAD + xg + 4 * k];
      s[0] = v.x; s[1] = v.y; s[2] = v.z; s[3] = v.w;
    }
  }
  __syncthreads();
#endif

  // Write transposed tile: flat f = x*32 + c, fully coalesced b128 stores.
  // LDS read banks: ((c+j)*129 + x) % 64 = (c + j + x) % 64 -> conflict-free.
#pragma unroll
  for (int k = 0; k < 4; ++k) {
    int f = (tid + k * 256) * 4;
    int c = f & (R_CH - 1);
    int x = f >> 5;
    float4 v;
    v.x = stage[(c + 0) * SPAD + x];
    v.y = stage[(c + 1) * SPAD + x];
    v.z = stage[(c + 2) * SPAD + x];
    v.w = stage[(c + 3) * SPAD + x];
    *(float4*)(out + f) = v;
  }
}

// ---------------------------------------------------------------------------
// Bilinear sample of one plane for this lane's channel.
// align_corners=true semantics; clamping x0<=510 with wx=x-x0 is exactly
// equivalent to the reference's clip (x in [0,511]).
// ---------------------------------------------------------------------------
template <bool TRANSPOSED>
__device__ __forceinline__ float vm_sample_plane(const float* __restrict__ base,
                                                 float gx, float gy, int lane)
{
  float x = (gx + 1.0f) * (0.5f * (RES - 1));
  float y = (gy + 1.0f) * (0.5f * (RES - 1));
  int x0 = (int)floorf(x);
  int y0 = (int)floorf(y);
  x0 = min(max(x0, 0), RES - 2);
  y0 = min(max(y0, 0), RES - 2);
  float wx = x - (float)x0;
  float wy = y - (float)y0;
  float v00, v01, v10, v11;
  if (TRANSPOSED) {
    const float* r0 = base + (size_t)(y0 * RES + x0) * R_CH + lane;
    v00 = r0[0];
    v01 = r0[R_CH];
    v10 = r0[RES * R_CH];
    v11 = r0[RES * R_CH + R_CH];
  } else {
    const float* r0 = base + (size_t)lane * CH_STRIDE + (size_t)y0 * RES + x0;
    v00 = r0[0];
    v01 = r0[1];
    v10 = r0[RES];
    v11 = r0[RES + 1];
  }
  float a = v00 + wx * (v01 - v00);
  float b = v10 + wx * (v11 - v10);
  return a + wy * (b - a);
}

// ---------------------------------------------------------------------------
// Gather kernel: lane == channel (wave32 <-> R=32). Each wave handles PPW
// points; 24 position floats loaded coalesced by lanes 0..23, broadcast by
// shuffle. Output store is one contiguous 128B line per point.
// ---------------------------------------------------------------------------
template <bool TRANSPOSED>
__global__ __launch_bounds__(256) void vm_gather(
    const float* __restrict__ pos,
    const float* __restrict__ p0, const float* __restrict__ p1,
    const float* __restrict__ p2,
    const float* __restrict__ l0, const float* __restrict__ l1,
    const float* __restrict__ l2,
    const float* __restrict__ tp,
    float* __restrict__ out)
{
  const int lane  = threadIdx.x & 31;
  const int wave  = blockIdx.x * 8 + (threadIdx.x >> 5);
  const int pbase = wave * PPW;

  // Per-channel line constants (see header comment).
  const float ls0 = 0.5f * (l0[lane * RES + 255] + l0[lane * RES + 256]);
  const float ls1 = 0.5f * (l1[lane * RES + 255] + l1[lane * RES + 256]);
  const float ls2 = 0.5f * (l2[lane * RES + 255] + l2[lane * RES + 256]);

  const float* b0 = TRANSPOSED ? tp : p0;
  const float* b1 = TRANSPOSED ? tp + PLANE_ELEMS : p1;
  const float* b2 = TRANSPOSED ? tp + 2 * (size_t)PLANE_ELEMS : p2;

  if (lane == 0) __builtin_prefetch(pos + (size_t)pbase * 3, 0, 1);

  // Coalesced position load: PPW*3 = 24 floats for this wave's points.
  float pv = 0.0f;
  if (lane < 3 * PPW) pv = pos[(size_t)pbase * 3 + lane];

#pragma unroll
  for (int i = 0; i < PPW; ++i) {
    const float px = __shfl(pv, 3 * i + 0, 32);
    const float py = __shfl(pv, 3 * i + 1, 32);
    const float pz = __shfl(pv, 3 * i + 2, 32);
    // plane_xy = [(0,1),(0,2),(1,2)] with gx=flat[:,a], gy=flat[:,b]
    float acc;
    acc  = vm_sample_plane<TRANSPOSED>(b0, px, py, lane) * ls0;
    acc += vm_sample_plane<TRANSPOSED>(b1, px, pz, lane) * ls1;
    acc += vm_sample_plane<TRANSPOSED>(b2, py, pz, lane) * ls2;
    out[(size_t)(pbase + i) * R_CH + lane] = acc;
  }
}

// ---------------------------------------------------------------------------
extern "C" void kernel_launch(void* const* d_in, const int* in_sizes, int n_in,
                              void* d_out, int out_size, void* d_ws,
                              size_t ws_size, hipStream_t stream)
{
  (void)in_sizes; (void)n_in; (void)out_size;
  // setup_inputs order: positions, plane0, line0, plane1, line1, plane2, line2
  const float* pos = (const float*)d_in[0];
  const float* p0  = (const float*)d_in[1];
  const float* l0  = (const float*)d_in[2];
  const float* p1  = (const float*)d_in[3];
  const float* l1  = (const float*)d_in[4];
  const float* p2  = (const float*)d_in[5];
  const float* l2  = (const float*)d_in[6];
  float* out = (float*)d_out;

  const int nblocks = NPTS / (8 * PPW);  // 8192 blocks x 256 threads

  if (ws_size >= WS_NEEDED) {
    float* tp = (float*)d_ws;
    dim3 tgrid(RES / XB, RES, 3);        // 4 x 512 x 3 tiles
    vm_transpose_planes<<<tgrid, 256, 0, stream>>>(p0, p1, p2, tp);
    vm_gather<true><<<nblocks, 256, 0, stream>>>(pos, p0, p1, p2, l0, l1, l2,
                                                 tp, out);
  } else {
    vm_gather<false><<<nblocks, 256, 0, stream>>>(pos, p0, p1, p2, l0, l1, l2,
                                                  nullptr, out);
  }
}